// RGCNBlockLayer_33380485825123
// MI455X (gfx1250) — compile-verified
//
#include <hip/hip_runtime.h>
#include <hip/hip_bf16.h>
#include <stdint.h>

// ---------------------------------------------------------------------------
// RGCN block-diagonal layer, fused per-edge formulation:
//   out[dst] += norm_e * (h[src] @ W_rel)      (W_rel block-diag, 8 x 32x32)
// Device counting-sort by relation, then 32-edge tiles processed with
// V_WMMA_F32_16X16X4_F32 (two 16-row sub-tiles share B operands in VGPRs).
// norm_e is folded into the A operand; pad rows carry norm=0 so their WMMA
// output is +/-0 and the scatter can be a fully unconditional atomic add.
// Gather uses the CDNA5 async global->LDS path (ASYNCcnt).
// ---------------------------------------------------------------------------

#define IN_FEAT   256
#define OUT_FEAT  256
#define NUM_RELS  8
#define NUM_BASES 8
#define SUB       32          // 32x32 sub-blocks
#define TILE_M    32          // edges per tile (2 WMMA sub-tiles of 16)
#define LDS_STRIDE 260        // 256 + 4 pad -> conflict-free column reads

typedef float v2f __attribute__((ext_vector_type(2)));
typedef float v8f __attribute__((ext_vector_type(8)));
typedef int   a_v4i __attribute__((vector_size(16)));   // async-copy payload

// ws layout (ints): [0..7] counts, [8..16] offsets, [17..25] tilePrefix,
//                   [26..33] cursor, [64..64+E) sorted ids
#define WS_COUNTS 0
#define WS_OFFS   8
#define WS_TILEP  17
#define WS_CURSOR 26
#define WS_SORTED 64

#if defined(__has_builtin)
#if __has_builtin(__builtin_amdgcn_global_load_async_to_lds_b128)
#define HAVE_ASYNC_LDS 1
#endif
#endif

// ---- zero output + workspace header ---------------------------------------
__global__ void rgcn_init_kernel(float* __restrict__ out, int n,
                                 int* __restrict__ ws) {
    int i = blockIdx.x * blockDim.x + threadIdx.x;
    int stride = gridDim.x * blockDim.x;
    for (int k = i; k < n; k += stride) out[k] = 0.0f;
    if (i < WS_SORTED) ws[i] = 0;
}

// ---- relation histogram ----------------------------------------------------
__global__ void rgcn_hist_kernel(const int* __restrict__ rel, int E,
                                 int* __restrict__ ws) {
    int i = blockIdx.x * blockDim.x + threadIdx.x;
    if (i < E) atomicAdd(&ws[WS_COUNTS + rel[i]], 1);
}

// ---- tiny exclusive scan (8 relations) ------------------------------------
__global__ void rgcn_scan_kernel(int* __restrict__ ws) {
    if (threadIdx.x == 0 && blockIdx.x == 0) {
        int o = 0, tp = 0;
        ws[WS_OFFS + 0]  = 0;
        ws[WS_TILEP + 0] = 0;
        for (int r = 0; r < NUM_RELS; ++r) {
            int c = ws[WS_COUNTS + r];
            ws[WS_CURSOR + r] = o;
            o += c;
            ws[WS_OFFS + r + 1] = o;
            tp += (c + TILE_M - 1) / TILE_M;
            ws[WS_TILEP + r + 1] = tp;
        }
    }
}

// ---- scatter edges into relation-sorted order ------------------------------
__global__ void rgcn_scatter_kernel(const int* __restrict__ rel, int E,
                                    int* __restrict__ ws) {
    int i = blockIdx.x * blockDim.x + threadIdx.x;
    if (i < E) {
        int p = atomicAdd(&ws[WS_CURSOR + rel[i]], 1);
        ws[WS_SORTED + p] = i;
    }
}

// ---- main WMMA tile kernel -------------------------------------------------
// Block = 256 threads = 8 waves; wave b owns base-block b (32x32 W in VGPRs).
// Per wave: two 16x32 A sub-tiles x shared W -> 32 v_wmma_f32_16x16x4_f32,
// then unconditional global_atomic_add_f32 scatter (pad rows add +/-0 to
// out[node 0], a numeric no-op).
__global__ __launch_bounds__(256)
void rgcn_tile_kernel(const float* __restrict__ h,
                      const float* __restrict__ weight,
                      const int*   __restrict__ edge_src,
                      const int*   __restrict__ edge_dst,
                      const float* __restrict__ edge_norm,
                      int*         __restrict__ ws,
                      float* __restrict__ out) {
    __shared__ __align__(16) float s_msg[TILE_M * LDS_STRIDE]; // raw h rows
    __shared__ int   s_src[TILE_M];
    __shared__ int   s_dst[TILE_M];
    __shared__ float s_norm[TILE_M];

    const int* offsets    = ws + WS_OFFS;
    const int* tilePrefix = ws + WS_TILEP;
    const int* sorted     = ws + WS_SORTED;

    int tile = blockIdx.x;
    if (tile >= tilePrefix[NUM_RELS]) return;      // uniform early-exit

    int r = 0;
    while (tile >= tilePrefix[r + 1]) ++r;         // <=8 scalar iterations
    int segStart = offsets[r] + (tile - tilePrefix[r]) * TILE_M;
    int cnt = offsets[r + 1] - segStart;
    if (cnt > TILE_M) cnt = TILE_M;

    int t = threadIdx.x;

    if (t < TILE_M) {
        if (t < cnt) {
            int e = sorted[segStart + t];
            s_src[t]  = edge_src[e];
            s_dst[t]  = edge_dst[e];
            s_norm[t] = edge_norm[e];
        } else {               // pad rows: zero norm -> zero WMMA output
            s_src[t]  = 0;
            s_dst[t]  = 0;
            s_norm[t] = 0.0f;
        }
    }
    __syncthreads();

    // ---- gather raw h[src] rows into LDS ----------------------------------
    {
        int row = t >> 3;          // 0..31
        int sub = t & 7;           // 8 threads per 1KB row
        int src = s_src[row];
#if HAVE_ASYNC_LDS
        // CDNA5 async global->LDS copy (ASYNCcnt), 16B per lane per op.
        // Constant GEPs fold into the instruction IOFFSET field.
        uint32_t ldsRow = (uint32_t)(uintptr_t)&s_msg[row * LDS_STRIDE];
        uint64_t gRow   = (uint64_t)(uintptr_t)(h + (size_t)src * IN_FEAT);
        __attribute__((address_space(1))) a_v4i* gp =
            (__attribute__((address_space(1))) a_v4i*)(uintptr_t)(gRow + sub * 16);
        __attribute__((address_space(3))) a_v4i* lp =
            (__attribute__((address_space(3))) a_v4i*)(uintptr_t)(ldsRow + sub * 16);
        #pragma unroll
        for (int j = 0; j < 8; ++j) {      // 8 chunks/thread, stride 128B
            __builtin_amdgcn_global_load_async_to_lds_b128(gp + j * 8, lp + j * 8, 0, 0);
        }
#if __has_builtin(__builtin_amdgcn_s_wait_asynccnt)
        __builtin_amdgcn_s_wait_asynccnt(0);
#else
        asm volatile("s_wait_asynccnt 0x0" ::: "memory");
#endif
#else
        const float4* hp = (const float4*)(h + (size_t)src * IN_FEAT);
        #pragma unroll
        for (int j = 0; j < 2; ++j) {
            int c4 = sub + j * 8;  // float4 index, 0..15
            float4 v = hp[c4];
            float* dp = &s_msg[row * LDS_STRIDE + c4 * 4];
            dp[0] = v.x; dp[1] = v.y; dp[2] = v.z; dp[3] = v.w;
        }
#endif
    }
    __syncthreads();

    int lane  = t & 31;
    int b     = t >> 5;                 // base block, uniform per wave
    int n     = lane & 15;              // B/D column within 16
    int khalf = lane >> 4;              // 0 -> K{0,1}, 1 -> K{2,3}

    // W_b[i][o] row-major (in x out), matches einsum "rbio"
    const float* W = weight + ((size_t)r * NUM_BASES + b) * (SUB * SUB);

    // B operands: Wb[nh][kk] = W[k..k+1][nh*16 + n] per ISA 4x16 layout
    v2f Wb[2][8];
    #pragma unroll
    for (int nh = 0; nh < 2; ++nh) {
        #pragma unroll
        for (int kk = 0; kk < 8; ++kk) {
            int krow = kk * 4 + khalf * 2;
            int col  = nh * 16 + n;
            Wb[nh][kk].x = W[krow * SUB + col];
            Wb[nh][kk].y = W[(krow + 1) * SUB + col];
        }
    }

    // per-lane norm scale for each A sub-tile row (row M = lane%16 (+16));
    // pad rows have norm==0 so their products vanish.
    float nrm0 = s_norm[n];
    float nrm1 = s_norm[16 + n];

    v8f acc[2][2] = {};   // [sub-tile s][column half nh]
    const float* arow0 = &s_msg[n * LDS_STRIDE + b * SUB + khalf * 2];
    const float* arow1 = arow0 + 16 * LDS_STRIDE;
    #pragma unroll
    for (int kk = 0; kk < 8; ++kk) {
        // A(16x4): lanes 0-15 K{0,1}, lanes 16-31 K{2,3}, M = lane%16
        v2f a0 = *(const v2f*)(arow0 + kk * 4);
        v2f a1 = *(const v2f*)(arow1 + kk * 4);
        a0.x *= nrm0; a0.y *= nrm0;
        a1.x *= nrm1; a1.y *= nrm1;
        acc[0][0] = __builtin_amdgcn_wmma_f32_16x16x4_f32(
                        false, a0, false, Wb[0][kk], (short)0, acc[0][0], false, false);
        acc[0][1] = __builtin_amdgcn_wmma_f32_16x16x4_f32(
                        false, a0, false, Wb[1][kk], (short)0, acc[0][1], false, false);
        acc[1][0] = __builtin_amdgcn_wmma_f32_16x16x4_f32(
                        false, a1, false, Wb[0][kk], (short)0, acc[1][0], false, false);
        acc[1][1] = __builtin_amdgcn_wmma_f32_16x16x4_f32(
                        false, a1, false, Wb[1][kk], (short)0, acc[1][1], false, false);
    }

    // ---- unconditional atomic scatter -------------------------------------
    // D layout: VGPR g -> M = g + (lane<16 ? 0 : 8), N = lane%16.
    // 32-bit unsigned element offsets (dst*256 + col < 2^31) -> saddr form.
    int mbase   = khalf * 8;
    uint32_t colBase = (uint32_t)(b * SUB + n);
    #pragma unroll
    for (int s = 0; s < 2; ++s) {
        #pragma unroll
        for (int g = 0; g < 8; ++g) {
            int mm = s * 16 + mbase + g;
            uint32_t off = (uint32_t)s_dst[mm] * OUT_FEAT + colBase;
            atomicAdd(out + off,      acc[s][0][g]);
            atomicAdd(out + off + 16, acc[s][1][g]);
        }
    }
}

// ---------------------------------------------------------------------------
extern "C" void kernel_launch(void* const* d_in, const int* in_sizes, int n_in,
                              void* d_out, int out_size, void* d_ws, size_t ws_size,
                              hipStream_t stream) {
    const float* h         = (const float*)d_in[0];
    const float* weight    = (const float*)d_in[1];
    const int*   edge_src  = (const int*)d_in[2];
    const int*   edge_dst  = (const int*)d_in[3];
    const int*   edge_rel  = (const int*)d_in[4];
    const float* edge_norm = (const float*)d_in[5];
    float* out = (float*)d_out;
    int*   ws  = (int*)d_ws;

    int E = in_sizes[2];

    rgcn_init_kernel<<<512, 256, 0, stream>>>(out, out_size, ws);
    rgcn_hist_kernel<<<(E + 255) / 256, 256, 0, stream>>>(edge_rel, E, ws);
    rgcn_scan_kernel<<<1, 32, 0, stream>>>(ws);
    rgcn_scatter_kernel<<<(E + 255) / 256, 256, 0, stream>>>(edge_rel, E, ws);

    int maxTiles = (E + TILE_M - 1) / TILE_M + NUM_RELS;  // extras exit early
    rgcn_tile_kernel<<<maxTiles, 256, 0, stream>>>(
        h, weight, edge_src, edge_dst, edge_norm, ws, out);
}